// CTCLayer_15899968930572
// MI455X (gfx1250) — compile-verified
//
#include <hip/hip_runtime.h>
#include <stdint.h>

// CTC forward loss, gfx1250 (MI455X).
// d_in[0] = y_true int32 [B,L], d_in[1] = y_pred float32 [B,T,C], d_out = float32 [B].
//
// One block per batch element. Thread s owns extended CTC state s (S = 2L+1 = 129).
// Probability rows stream HBM -> LDS through the CDNA5 async path
// (global_load_async_to_lds_b128, ASYNCcnt) with a 4-deep ring and 2 rows in
// flight; the whole recursion runs in log2 space (v_exp_f32/v_log_f32 are
// native base-2), converted to natural log once at the end.

#define CTC_EPS  1e-7f
#define CTC_NEG  -1e30f
#define CTC_LN2  0.69314718055994530942f

constexpr int kC       = 256;   // classes (blank = kC-1)
constexpr int kThreads = 160;   // 5 wave32s: 129 state lanes; waves 0-1 are loaders
constexpr int kRing    = 4;     // row ring depth (2 rows in flight)
constexpr int kMaxS    = 132;   // 2*64+1 states + 2-entry NEG pad, rounded up

// native base-2 transcendentals (v_exp_f32 / v_log_f32)
__device__ __forceinline__ float fexp2(float x) { return __builtin_amdgcn_exp2f(x); }
__device__ __forceinline__ float flog2(float x) { return __builtin_amdgcn_logf(x); }

__device__ __forceinline__ void async_row_load_b128(uint32_t lds_byte_off,
                                                    const float* gaddr) {
    asm volatile("global_load_async_to_lds_b128 %0, %1, off"
                 :: "v"(lds_byte_off), "v"(gaddr)
                 : "memory");
}

__device__ __forceinline__ void wait_async_le2() {
    asm volatile("s_wait_asynccnt 0x2" ::: "memory");
}
__device__ __forceinline__ void wait_async_le1() {
    asm volatile("s_wait_asynccnt 0x1" ::: "memory");
}
__device__ __forceinline__ void wait_async_le0() {
    asm volatile("s_wait_asynccnt 0x0" ::: "memory");
}

__global__ __launch_bounds__(kThreads)
void ctc_forward_kernel(const int* __restrict__ y_true,
                        const float* __restrict__ y_pred,
                        float* __restrict__ out,
                        int T, int L) {
    const int b     = blockIdx.x;
    const int tid   = threadIdx.x;
    const int S     = 2 * L + 1;
    const int blank = kC - 1;

    __shared__ float rbuf[kRing][kC];  // ring of probability rows
    __shared__ float alpha[2][kMaxS];  // double-buffered alpha, NEG pad at [0..1]

    // ---- per-thread static CTC state (thread s owns extended state s) ----
    const int  s      = tid;
    const bool active = (s < S);
    int  ext_s   = blank;
    bool skip_ok = false;
    if (active && (s & 1)) {
        const int li = s >> 1;                       // (s-1)/2 for odd s
        ext_s = y_true[(size_t)b * L + li];
        // skip allowed iff label differs from label two states back;
        // for s==1 the pad entry alpha[...][1] == NEG keeps it harmless.
        skip_ok = (li == 0) ? true
                            : (ext_s != y_true[(size_t)b * L + li - 1]);
    }

    // NEG pad for the s-1 / s-2 shifted reads
    if (tid < 2) { alpha[0][tid] = CTC_NEG; alpha[1][tid] = CTC_NEG; }

    const float* rowbase = y_pred + (size_t)b * T * kC;

    // LDS byte offsets of this thread's 16B slot in each ring buffer.
    uint32_t ldsoff[kRing];
    if (tid < 64) {
#pragma unroll
        for (int r = 0; r < kRing; ++r)
            ldsoff[r] = (uint32_t)(uintptr_t)(&rbuf[r][tid * 4]);
        // prologue: rows 0 and 1 in flight
        async_row_load_b128(ldsoff[0], rowbase + tid * 4);
        if (T > 1)
            async_row_load_b128(ldsoff[1], rowbase + (size_t)kC + tid * 4);
    }

    for (int t = 0; t < T; ++t) {
        // Keep 2 rows in flight; wait until row t has landed in LDS.
        if (t + 2 < T) {
            if (tid < 64)
                async_row_load_b128(ldsoff[(t + 2) & (kRing - 1)],
                                    rowbase + (size_t)(t + 2) * kC + tid * 4);
            wait_async_le2();
        } else if (t + 1 < T) {
            wait_async_le1();
        } else {
            wait_async_le0();
        }

        // Single barrier: makes row t + alpha(t-1) visible, and orders the
        // rbuf slot reuse (distance 4) and alpha buffer reuse (distance 2).
        __syncthreads();

        if (active) {
            // log2-space emission prob
            const float lp = flog2(rbuf[t & (kRing - 1)][ext_s] + CTC_EPS);
            float nv;
            if (t == 0) {
                nv = (s <= 1) ? lp : CTC_NEG;
            } else {
                const float* R = alpha[(t & 1) ^ 1];
                const float a1 = R[s + 2];                     // alpha[s]
                const float a2 = R[s + 1];                     // alpha[s-1]
                const float a3 = skip_ok ? R[s] : CTC_NEG;     // alpha[s-2]
                const float m  = fmaxf(a1, fmaxf(a2, a3));
                nv = m + flog2(fexp2(a1 - m) + fexp2(a2 - m) +
                               fexp2(a3 - m)) + lp;
            }
            alpha[t & 1][s + 2] = nv;
        }
    }

    __syncthreads();
    if (tid == 0) {
        const float* A = alpha[(T - 1) & 1];
        const float x = A[S + 1];   // alpha[S-1]
        const float y = A[S];       // alpha[S-2]
        const float m = fmaxf(x, y);
        // back to natural log once at the end
        out[b] = -CTC_LN2 * (m + flog2(fexp2(x - m) + fexp2(y - m)));
    }
}

extern "C" void kernel_launch(void* const* d_in, const int* in_sizes, int n_in,
                              void* d_out, int out_size, void* d_ws, size_t ws_size,
                              hipStream_t stream) {
    const int*   y_true = (const int*)d_in[0];
    const float* y_pred = (const float*)d_in[1];
    float*       out    = (float*)d_out;

    const int B = out_size;                       // 512
    const int L = in_sizes[0] / B;                // 64
    const int T = in_sizes[1] / (B * kC);         // 1024

    ctc_forward_kernel<<<B, kThreads, 0, stream>>>(y_true, y_pred, out, T, L);
}